// GATLayer_223338300196
// MI455X (gfx1250) — compile-verified
//
#include <hip/hip_runtime.h>
#include <hip/hip_bf16.h>
#include <math.h>

typedef __attribute__((ext_vector_type(2))) float v2f;
typedef __attribute__((ext_vector_type(8))) float v8f;

#define GAT_INDIM 128
#define GAT_FDIM  128   // N_HEADS * OUT_DIM
#define GAT_HEADS 4
#define GAT_ODIM  32
#define NEG_SLOPE 0.2f
#define PROJ_ROWS 64    // node rows per block in the WMMA projection

// ---------------------------------------------------------------------------
// Kernel 1: feat = h @ W via V_WMMA_F32_16X16X4_F32 (full f32 precision).
// Block = 256 threads (8 waves). Block owns 64 node rows; wave w owns output
// columns [16w, 16w+16) and iterates 4 row tiles (4 accumulators) so W is
// staged once per 64 rows instead of per 16 (4x less L2 traffic on W).
// All LDS staging is float4-vectorized (b128 loads/stores).
// ---------------------------------------------------------------------------
__global__ void __launch_bounds__(256)
gat_proj_wmma(const float* __restrict__ h, const float* __restrict__ W,
              float* __restrict__ feat, int n_nodes) {
  __shared__ float sW[64 * GAT_FDIM];        // 32 KB: K-slab of 64 rows of W
  __shared__ float sA[PROJ_ROWS * GAT_INDIM]; // 32 KB: 64 node rows

  const int t      = threadIdx.x;
  const int m_base = blockIdx.x * PROJ_ROWS;
  const int wv     = t >> 5;       // wave id 0..7 -> output col tile
  const int lane   = t & 31;
  const int n0     = wv * 16;
  const int hl     = lane >> 4;    // lane half (0/1)
  const int l15    = lane & 15;

  // Stage the 64 A rows once (float4, coalesced, guarded per row).
  for (int i = t; i < PROJ_ROWS * 32; i += 256) {  // 32 float4 per row
    const int r = m_base + (i >> 5);
    float4 v = make_float4(0.f, 0.f, 0.f, 0.f);
    if (r < n_nodes) v = ((const float4*)h)[(size_t)r * 32 + (i & 31)];
    ((float4*)sA)[i] = v;
  }

  v8f acc[4] = {};  // row tiles rt: rows [m_base + 16*rt, +16)
  for (int kk = 0; kk < GAT_INDIM; kk += 64) {
    __syncthreads();
    // Stage 64 K-rows of W (float4, coalesced).
    for (int i = t; i < 64 * 32; i += 256)
      ((float4*)sW)[i] = ((const float4*)W)[(size_t)kk * 32 + i];
    __syncthreads();

    for (int k0 = 0; k0 < 64; k0 += 4) {
      const int ka = k0 + 2 * hl;  // this lane-half's K within the slab
      // B 4x16 f32: lanes hold N=l15; VGPR0/1 = rows K, K+1 (half -> +2)
      v2f b;
      b.x = sW[ka * GAT_FDIM + n0 + l15];
      b.y = sW[(ka + 1) * GAT_FDIM + n0 + l15];
      #pragma unroll
      for (int rt = 0; rt < 4; ++rt) {
        // A 16x4 f32: lanes 0-15 hold M; VGPR0/1 = K, K+1 (half -> +2)
        v2f a;
        a.x = sA[(rt * 16 + l15) * GAT_INDIM + ka];
        a.y = sA[(rt * 16 + l15) * GAT_INDIM + ka + 1];
        acc[rt] = __builtin_amdgcn_wmma_f32_16x16x4_f32(
            /*neg_a=*/false, a, /*neg_b=*/false, b,
            /*c_mod=*/(short)0, acc[rt], /*reuse_a=*/false, /*reuse_b=*/false);
      }
    }
  }

  // D 16x16 f32 layout: VGPR v -> row M = v + 8*half, col N = l15.
  #pragma unroll
  for (int rt = 0; rt < 4; ++rt) {
    #pragma unroll
    for (int v = 0; v < 8; ++v) {
      const int row = m_base + rt * 16 + v + 8 * hl;
      if (row < n_nodes)
        feat[(size_t)row * GAT_FDIM + n0 + l15] = acc[rt][v];
    }
  }
}

// ---------------------------------------------------------------------------
// Kernel 2: el[n,h] = sum_d feat[n,h,d]*attn_l[h,d]; same for er.
// One 128-thread block per node; wave w == head w; shuffle reduction.
// ---------------------------------------------------------------------------
__global__ void __launch_bounds__(128)
gat_attn_logits(const float* __restrict__ feat,
                const float* __restrict__ attn_l,
                const float* __restrict__ attn_r,
                float* __restrict__ el, float* __restrict__ er) {
  const int n = blockIdx.x;
  const int t = threadIdx.x;
  const float f = feat[(size_t)n * GAT_FDIM + t];
  float vl = f * attn_l[t];
  float vr = f * attn_r[t];
  #pragma unroll
  for (int off = 16; off > 0; off >>= 1) {
    vl += __shfl_down(vl, off, 32);
    vr += __shfl_down(vr, off, 32);
  }
  if ((t & 31) == 0) {
    el[n * GAT_HEADS + (t >> 5)] = vl;
    er[n * GAT_HEADS + (t >> 5)] = vr;
  }
}

// ---------------------------------------------------------------------------
// Kernel 3: per-destination segment softmax + weighted aggregation + ReLU.
// One 128-thread block per destination node (dst is sorted). thread t covers
// output feature f = t (head = t>>5, d = t&31). Edge scores recomputed on the
// fly (el gather + er broadcast), weights staged through LDS in 32-edge
// chunks so feat[src] reads are 512B-coalesced per edge. feat (51.2 MB) is
// L2-resident (192 MB L2), so the 819 MB gather stream stays on-chip.
// ---------------------------------------------------------------------------
__global__ void __launch_bounds__(128)
gat_aggregate(const float* __restrict__ feat, const float* __restrict__ el,
              const float* __restrict__ er, const int* __restrict__ src,
              const int* __restrict__ dst, const float* __restrict__ bias,
              float* __restrict__ out, int n_edges) {
  __shared__ int   s_range[2];
  __shared__ float s_er[GAT_HEADS];
  __shared__ float s_max[GAT_HEADS];
  __shared__ float s_den[GAT_HEADS];
  __shared__ int   s_src[32];
  __shared__ float s_w[GAT_HEADS][32];

  const int n    = blockIdx.x;
  const int t    = threadIdx.x;
  const int head = t >> 5;
  const int lane = t & 31;

  if (t == 0) {
    int lo = 0, hi = n_edges;                 // lower_bound(dst, n)
    while (lo < hi) { int mid = (lo + hi) >> 1; if (dst[mid] < n) lo = mid + 1; else hi = mid; }
    s_range[0] = lo;
    int lo2 = lo; hi = n_edges;               // upper_bound(dst, n)
    while (lo2 < hi) { int mid = (lo2 + hi) >> 1; if (dst[mid] <= n) lo2 = mid + 1; else hi = mid; }
    s_range[1] = lo2;
  }
  if (t < GAT_HEADS) s_er[t] = er[n * GAT_HEADS + t];
  __syncthreads();

  const int s = s_range[0], e_end = s_range[1];
  const float ern = s_er[head];

  // Pass 1: per-head max (wave `head` covers its head's edges).
  float m = -3.402823e38f;
  for (int e = s + lane; e < e_end; e += 32) {
    float sc = el[src[e] * GAT_HEADS + head] + ern;
    sc = sc > 0.0f ? sc : NEG_SLOPE * sc;
    m = fmaxf(m, sc);
  }
  #pragma unroll
  for (int off = 16; off > 0; off >>= 1) m = fmaxf(m, __shfl_xor(m, off, 32));
  if (lane == 0) s_max[head] = m;
  __syncthreads();

  // Pass 2: per-head denom.
  const float mm = s_max[head];
  float z = 0.0f;
  for (int e = s + lane; e < e_end; e += 32) {
    float sc = el[src[e] * GAT_HEADS + head] + ern;
    sc = sc > 0.0f ? sc : NEG_SLOPE * sc;
    z += expf(sc - mm);
  }
  #pragma unroll
  for (int off = 16; off > 0; off >>= 1) z += __shfl_xor(z, off, 32);
  if (lane == 0) s_den[head] = z;
  __syncthreads();

  const float invz = (e_end > s) ? 1.0f / s_den[head] : 0.0f;

  // Pass 3: chunked weighted aggregation (fixed order -> deterministic).
  float acc = 0.0f;
  for (int base = s; base < e_end; base += 32) {
    const int cnt = min(32, e_end - base);
    __syncthreads();  // previous chunk fully consumed before overwrite
    if (lane < cnt) {
      const int se = src[base + lane];
      if (head == 0) s_src[lane] = se;
      float sc = el[se * GAT_HEADS + head] + ern;
      sc = sc > 0.0f ? sc : NEG_SLOPE * sc;
      s_w[head][lane] = expf(sc - mm) * invz;
    }
    __syncthreads();
    for (int j = 0; j < cnt; ++j)
      acc += s_w[head][j] * feat[(size_t)s_src[j] * GAT_FDIM + t];
  }

  out[(size_t)n * GAT_FDIM + t] = fmaxf(acc + bias[t], 0.0f);
}

// ---------------------------------------------------------------------------
// Launch
// ---------------------------------------------------------------------------
extern "C" void kernel_launch(void* const* d_in, const int* in_sizes, int n_in,
                              void* d_out, int out_size, void* d_ws, size_t ws_size,
                              hipStream_t stream) {
  const float* h      = (const float*)d_in[0];
  const int*   src    = (const int*)  d_in[1];
  const int*   dst    = (const int*)  d_in[2];
  const float* W      = (const float*)d_in[3];
  const float* attn_l = (const float*)d_in[4];
  const float* attn_r = (const float*)d_in[5];
  const float* bias   = (const float*)d_in[6];

  const int n_nodes = in_sizes[0] / GAT_INDIM;
  const int n_edges = in_sizes[1];

  float* feat = (float*)d_ws;                          // n_nodes*128 floats
  float* el   = feat + (size_t)n_nodes * GAT_FDIM;     // n_nodes*4
  float* er   = el   + (size_t)n_nodes * GAT_HEADS;    // n_nodes*4

  float* out = (float*)d_out;

  const int proj_blocks = (n_nodes + PROJ_ROWS - 1) / PROJ_ROWS;
  gat_proj_wmma<<<proj_blocks, 256, 0, stream>>>(h, W, feat, n_nodes);
  gat_attn_logits<<<n_nodes, 128, 0, stream>>>(feat, attn_l, attn_r, el, er);
  gat_aggregate<<<n_nodes, 128, 0, stream>>>(feat, el, er, src, dst, bias,
                                             out, n_edges);
}